// Attention2D_89550068122322
// MI455X (gfx1250) — compile-verified
//
#include <hip/hip_runtime.h>
#include <math.h>

// Attention2D for gfx1250 (MI455X): fp32 WMMA (v_wmma_f32_16x16x4_f32) +
// Tensor Data Mover staging of GEMM B-panels into LDS.
// WMMA operand layouts per CDNA5 ISA 7.12.2 (wave32):
//   A (16x4):  lane l: M = l&15 ; v0 = A[M][k0+2*(l>>4)], v1 = A[M][k0+2*(l>>4)+1]
//   B (4x16):  lane l: N = l&15 ; v0 = B[k0+2*(l>>4)][N], v1 = B[k0+2*(l>>4)+1][N]
//   C/D 16x16: vgpr i: lanes0-15 -> M=i, lanes16-31 -> M=i+8 ; N = l&15

typedef float v2f __attribute__((ext_vector_type(2)));
typedef float v8f __attribute__((ext_vector_type(8)));
typedef unsigned int u32x4 __attribute__((ext_vector_type(4)));
typedef int i32x4 __attribute__((ext_vector_type(4)));
typedef int i32x8 __attribute__((ext_vector_type(8)));

#define DIMC   256
#define HEADS  8
#define HD     32
#define NTOK   2304      // 48*48
#define NT16   (NTOK/16) // 144
#define SCALE  0.17677669529663687f  // 32^-0.5
#define PADTOK 36
#define KB     32        // k-panel staged per TDM transfer
#define LDSW   17        // LDS row stride in floats (16 data + 1 TDM pad dword)

__device__ __forceinline__ v8f wmma4(v2f a, v2f b, v8f c) {
  return __builtin_amdgcn_wmma_f32_16x16x4_f32(
      /*neg_a=*/false, a, /*neg_b=*/false, b,
      /*c_mod=*/(short)0, c, /*reuse_a=*/false, /*reuse_b=*/false);
}

// TDM 2D tile load: KB rows x 16 floats from row-major src (row stride = NTOK
// floats) into LDS with 1-dword pad every 16 dwords (=> LDSW-float row stride).
// D# per CDNA5 ISA ch.8: group0 = {count|flags, lds_addr, gaddr_lo, gaddr_hi|type},
// group1 = {mask/dsize/pad, abar, tdim0/1, tile dims, strides}, groups 2/3 = 0 (2D).
// This toolchain's builtin takes 6 args (g0, g1, g2, g3, extra group, cpol).
__device__ __forceinline__ void tdm_load_panel(const float* src, unsigned lds_off) {
  const unsigned long long ga = (unsigned long long)(size_t)src;
  u32x4 g0;
  g0[0] = 1u;                                           // count=1, user mode
  g0[1] = lds_off;                                      // LDS byte address
  g0[2] = (unsigned)ga;                                 // global_addr[31:0]
  g0[3] = (unsigned)(ga >> 32) | 0x80000000u;           // global_addr[56:32] | type=2
  i32x8 g1;
  g1[0] = 0x00D20000;   // data_size=4B(code2)<<16 | pad_enable<<20 | pad_interval=3(16dw)<<22
  g1[1] = (int)(NTOK << 16);          // tensor_dim0[15:0] in [63:48]  (2304)
  g1[2] = (int)(DIMC << 16);          // tensor_dim0[31:16]=0 | tensor_dim1[15:0] (256)
  g1[3] = (int)(16 << 16);            // tensor_dim1[31:16]=0 | tile_dim0=16
  g1[4] = KB;                         // tile_dim1=KB | tile_dim2=0
  g1[5] = NTOK;                       // tensor_dim0_stride[31:0]
  g1[6] = 0;                          // stride0[47:32] | tensor_dim1_stride[15:0]
  g1[7] = 0;                          // tensor_dim1_stride[47:16]
  i32x4 z4 = {0, 0, 0, 0};
  i32x8 z8 = {0, 0, 0, 0, 0, 0, 0, 0};
  __builtin_amdgcn_tensor_load_to_lds(g0, g1, z4, z4, z8, 0);
}

// ---------------------------------------------------------------------------
// Kernel 1: QKV projections.  Y = W (256x256) @ X (256x2304) + b
// grid = (144, 2, 3); block = 256 threads (8 waves). Each block computes a
// 128x16 tile; the X k-panel is TDM-staged into LDS once and shared by all
// 8 waves (wave w owns rows m0+16w..+15). z selects {Q, K, V}; V is stored
// transposed (N x C) so PV-stage A-operand loads in attention are coalesced.
// ---------------------------------------------------------------------------
__global__ __launch_bounds__(256) void qkv_gemm_f32wmma(
    const float* __restrict__ X,
    const float* __restrict__ Wq, const float* __restrict__ bq,
    const float* __restrict__ Wk, const float* __restrict__ bk,
    const float* __restrict__ Wv, const float* __restrict__ bv,
    float* __restrict__ Q, float* __restrict__ K, float* __restrict__ Vt)
{
  __shared__ float xs[KB * LDSW];
  const int tid  = threadIdx.x;
  const int l    = tid & 31;
  const int li   = l & 15;
  const int half = l >> 4;
  const int wave = tid >> 5;
  const int n0   = blockIdx.x * 16;
  const int m0   = blockIdx.y * 128 + wave * 16;
  const int which = blockIdx.z;

  const float* W    = (which == 0) ? Wq : (which == 1) ? Wk : Wv;
  const float* bias = (which == 0) ? bq : (which == 1) ? bk : bv;
  const unsigned lds_off = (unsigned)(size_t)(&xs[0]);   // LDS byte offset

  const int m = m0 + li;
  v8f c = {};
  for (int k0 = 0; k0 < DIMC; k0 += KB) {
    __syncthreads();                       // previous panel fully consumed
    if (tid < 32) {                        // wave 0 drives the TDM
      tdm_load_panel(X + (size_t)k0 * NTOK + n0, lds_off);
      __builtin_amdgcn_s_wait_tensorcnt(0);
    }
    __syncthreads();                       // panel visible to all waves
#pragma unroll
    for (int kc = 0; kc < KB; kc += 4) {
      const int r = kc + 2 * half;
      v2f a = *(const v2f*)(W + m * DIMC + k0 + r);  // 8B-aligned (r even)
      v2f b;
      b.x = xs[(r    ) * LDSW + li];
      b.y = xs[(r + 1) * LDSW + li];
      c = wmma4(a, b, c);
    }
  }

  if (which < 2) {
    float* Y = (which == 0) ? Q : K;
#pragma unroll
    for (int i = 0; i < 8; ++i) {
      const int mm = m0 + i + 8 * half;
      Y[mm * NTOK + n0 + li] = c[i] + bias[mm];
    }
  } else {
#pragma unroll
    for (int i = 0; i < 8; ++i) {
      const int mm = m0 + i + 8 * half;
      Vt[(n0 + li) * DIMC + mm] = c[i] + bias[mm];
    }
  }
}

// ---------------------------------------------------------------------------
// Kernel 2: fused flash attention with relative-position bias.
// grid = (72, 8): one wave per (2 x 16-query tiles, head).  K/V tile operands
// are loaded once and reused by both query tiles (2x WMMA per byte loaded).
// S^T = K^T Q so each lane owns ONE query column -> softmax reduction is
// 8 in-lane values + one shfl_xor(16).  Online softmax; O += V @ P^T.
// ---------------------------------------------------------------------------
__global__ __launch_bounds__(32) void flash_attn_f32wmma(
    const float* __restrict__ Q, const float* __restrict__ K,
    const float* __restrict__ Vt, const float* __restrict__ emb,
    const int* __restrict__ tokens, float* __restrict__ O)
{
  const int l    = threadIdx.x;
  const int li   = l & 15;
  const int half = l >> 4;
  const int n0   = blockIdx.x * 32;
  const int h    = blockIdx.y;

  const float* Qh = Q + h * HD * NTOK;
  const float* Kh = K + h * HD * NTOK;

  // Preload both Q tiles as B operands (8 chunks of K=4 over d=0..31)
  v2f bq[2][8];
#pragma unroll
  for (int t = 0; t < 2; ++t)
#pragma unroll
    for (int cc = 0; cc < 8; ++cc) {
      const int d = 4 * cc + 2 * half;
      bq[t][cc].x = Qh[(d    ) * NTOK + n0 + 16 * t + li];
      bq[t][cc].y = Qh[(d + 1) * NTOK + n0 + 16 * t + li];
    }

  const int* tokrow[2] = { tokens + (long)(n0      + li) * NTOK,
                           tokens + (long)(n0 + 16 + li) * NTOK };

  float m_run[2] = { -3.0e38f, -3.0e38f };
  float s_run[2] = { 0.f, 0.f };
  v8f o0[2] = {}, o1[2] = {};

  for (int m0 = 0; m0 < NTOK; m0 += 16) {
    // ---- S^T tiles = K^T Q : K operand loaded once, 2 WMMAs per chunk
    v8f s[2] = {};
#pragma unroll
    for (int cc = 0; cc < 8; ++cc) {
      const int d = 4 * cc + 2 * half;
      v2f a;
      a.x = Kh[(d    ) * NTOK + m0 + li];
      a.y = Kh[(d + 1) * NTOK + m0 + li];
      s[0] = wmma4(a, bq[0][cc], s[0]);
      s[1] = wmma4(a, bq[1][cc], s[1]);
    }

    v2f pb[2][4];
#pragma unroll
    for (int t = 0; t < 2; ++t) {
      // ---- scale + relative-position bias, local max
      float p[8];
      float lmax = -3.0e38f;
#pragma unroll
      for (int i = 0; i < 8; ++i) {
        const int mkey = m0 + i + 8 * half;        // key row held by this vgpr
        const int tok  = tokrow[t][mkey];
        const float bv = (tok == PADTOK) ? 0.f : emb[tok * HEADS + h];
        const float sv = s[t][i] * SCALE + bv;
        p[i] = sv;
        lmax = fmaxf(lmax, sv);
      }
      lmax = fmaxf(lmax, __shfl_xor(lmax, 16, 32));
      const float new_m = fmaxf(m_run[t], lmax);
      const float corr  = __expf(m_run[t] - new_m);

      float lsum = 0.f;
#pragma unroll
      for (int i = 0; i < 8; ++i) {
        p[i] = __expf(p[i] - new_m);
        lsum += p[i];
      }
      lsum += __shfl_xor(lsum, 16, 32);
      s_run[t] = s_run[t] * corr + lsum;
      m_run[t] = new_m;
#pragma unroll
      for (int i = 0; i < 8; ++i) { o0[t][i] *= corr; o1[t][i] *= corr; }

      // ---- convert P^T (C-layout) into B-operand chunks (keys 4c..4c+3)
      const float sh0 = __shfl_xor(p[0], 16, 32);
      const float sh1 = __shfl_xor(p[1], 16, 32);
      const float sh2 = __shfl_xor(p[2], 16, 32);
      const float sh3 = __shfl_xor(p[3], 16, 32);
      const float sh4 = __shfl_xor(p[4], 16, 32);
      const float sh5 = __shfl_xor(p[5], 16, 32);
      const float sh6 = __shfl_xor(p[6], 16, 32);
      const float sh7 = __shfl_xor(p[7], 16, 32);
      pb[t][0].x = half ? sh2 : p[0];  pb[t][0].y = half ? sh3 : p[1]; // keys 0..3
      pb[t][1].x = half ? sh6 : p[4];  pb[t][1].y = half ? sh7 : p[5]; // keys 4..7
      pb[t][2].x = half ? p[2] : sh0;  pb[t][2].y = half ? p[3] : sh1; // keys 8..11
      pb[t][3].x = half ? p[6] : sh4;  pb[t][3].y = half ? p[7] : sh5; // keys 12..15
    }

    // ---- O(d x queries) += V @ P^T : V operand loaded once, 4 WMMAs per chunk
#pragma unroll
    for (int cc = 0; cc < 4; ++cc) {
      const int kr = m0 + 4 * cc + 2 * half;       // key rows kr, kr+1
      v2f a0, a1;
      a0.x = Vt[(kr    ) * DIMC + h * HD + li];
      a0.y = Vt[(kr + 1) * DIMC + h * HD + li];
      a1.x = Vt[(kr    ) * DIMC + h * HD + 16 + li];
      a1.y = Vt[(kr + 1) * DIMC + h * HD + 16 + li];
      o0[0] = wmma4(a0, pb[0][cc], o0[0]);
      o1[0] = wmma4(a1, pb[0][cc], o1[0]);
      o0[1] = wmma4(a0, pb[1][cc], o0[1]);
      o1[1] = wmma4(a1, pb[1][cc], o1[1]);
    }
  }

#pragma unroll
  for (int t = 0; t < 2; ++t) {
    const float inv = 1.f / s_run[t];
#pragma unroll
    for (int i = 0; i < 8; ++i) {
      const int d = i + 8 * half;
      O[(h * HD + d     ) * NTOK + n0 + 16 * t + li] = o0[t][i] * inv;
      O[(h * HD + d + 16) * NTOK + n0 + 16 * t + li] = o1[t][i] * inv;
    }
  }
}

// ---------------------------------------------------------------------------
// Kernel 3: output projection  out = Wp @ A + bp, written in (1,256,48,48).
// Same TDM-staged structure as kernel 1. grid = (144, 2), 256 threads.
// ---------------------------------------------------------------------------
__global__ __launch_bounds__(256) void proj_gemm_f32wmma(
    const float* __restrict__ A, const float* __restrict__ W,
    const float* __restrict__ bias, float* __restrict__ Y)
{
  __shared__ float xs[KB * LDSW];
  const int tid  = threadIdx.x;
  const int l    = tid & 31;
  const int li   = l & 15;
  const int half = l >> 4;
  const int wave = tid >> 5;
  const int n0   = blockIdx.x * 16;
  const int m0   = blockIdx.y * 128 + wave * 16;
  const unsigned lds_off = (unsigned)(size_t)(&xs[0]);

  const int m = m0 + li;
  v8f c = {};
  for (int k0 = 0; k0 < DIMC; k0 += KB) {
    __syncthreads();
    if (tid < 32) {
      tdm_load_panel(A + (size_t)k0 * NTOK + n0, lds_off);
      __builtin_amdgcn_s_wait_tensorcnt(0);
    }
    __syncthreads();
#pragma unroll
    for (int kc = 0; kc < KB; kc += 4) {
      const int r = kc + 2 * half;
      v2f a = *(const v2f*)(W + m * DIMC + k0 + r);
      v2f b;
      b.x = xs[(r    ) * LDSW + li];
      b.y = xs[(r + 1) * LDSW + li];
      c = wmma4(a, b, c);
    }
  }
#pragma unroll
  for (int i = 0; i < 8; ++i) {
    const int mm = m0 + i + 8 * half;
    Y[mm * NTOK + n0 + li] = c[i] + bias[mm];
  }
}

extern "C" void kernel_launch(void* const* d_in, const int* in_sizes, int n_in,
                              void* d_out, int out_size, void* d_ws, size_t ws_size,
                              hipStream_t stream) {
  const float* x      = (const float*)d_in[0];
  const float* Wq     = (const float*)d_in[1];
  const float* bq     = (const float*)d_in[2];
  const float* Wk     = (const float*)d_in[3];
  const float* bk     = (const float*)d_in[4];
  const float* Wv     = (const float*)d_in[5];
  const float* bv     = (const float*)d_in[6];
  const float* Wp     = (const float*)d_in[7];
  const float* bp     = (const float*)d_in[8];
  const float* emb    = (const float*)d_in[9];
  const int*   tokens = (const int*)d_in[10];
  float* out = (float*)d_out;

  float* ws = (float*)d_ws;
  float* Q  = ws;                        // 256 x 2304
  float* K  = Q  + DIMC * NTOK;          // 256 x 2304
  float* Vt = K  + DIMC * NTOK;          // 2304 x 256 (transposed V)
  float* AO = Vt + NTOK * DIMC;          // 256 x 2304 (pre-projection attn out)

  qkv_gemm_f32wmma<<<dim3(NT16, DIMC / 128, 3), 256, 0, stream>>>(
      x, Wq, bq, Wk, bk, Wv, bv, Q, K, Vt);
  flash_attn_f32wmma<<<dim3(NT16 / 2, HEADS), 32, 0, stream>>>(
      Q, K, Vt, emb, tokens, AO);
  proj_gemm_f32wmma<<<dim3(NT16, DIMC / 128), 256, 0, stream>>>(
      AO, Wp, bp, out);
}